// ContextualAttention_44573170598865
// MI455X (gfx1250) — compile-verified
//
#include <hip/hip_runtime.h>
#include <cstdint>

typedef __attribute__((ext_vector_type(2))) float v2f;
typedef __attribute__((ext_vector_type(8))) float v8f;
typedef __attribute__((ext_vector_type(4))) int   i4v;

#define B_N 8
#define C_N 16
#define H_N 128
#define W_N 128
#define HD  64          // downsampled spatial
#define P_TOT 4096      // 64*64 patch positions
#define KDIM 144        // 16*3*3
#define KP2  148        // LDS row pad: 592B rows (16B aligned), conflict-free operand reads
#define CP   256        // 16*4*4 recon columns
#define NCH  (P_TOT / 16)
#define SM_SCALE 10.0f
#define NORM_EPS 1e-4f

// -------- async global->LDS copy (CDNA5), with safe fallback --------
#if defined(__AMDGCN__) && \
    __has_builtin(__builtin_amdgcn_global_load_async_to_lds_b128) && \
    __has_builtin(__builtin_amdgcn_s_wait_asynccnt)
#define HAVE_ASYNC 1
#else
#define HAVE_ASYNC 0
#endif

#if HAVE_ASYNC
typedef __attribute__((address_space(1))) i4v* glob_i4p;
typedef __attribute__((address_space(3))) i4v* lds_i4p;
#endif

__device__ __forceinline__ void copy16_to_lds(const float* gp, float* lp) {
#if HAVE_ASYNC
  __builtin_amdgcn_global_load_async_to_lds_b128(
      (glob_i4p)(uintptr_t)gp,
      (lds_i4p)(uintptr_t)lp,
      0, 0);
#else
  lp[0] = gp[0]; lp[1] = gp[1]; lp[2] = gp[2]; lp[3] = gp[3];
#endif
}

__device__ __forceinline__ void wait_async_done() {
#if HAVE_ASYNC
  __builtin_amdgcn_s_wait_asynccnt(0);
#endif
}

// ---------------- kernel 1: patch extraction / normalization ----------------
__global__ __launch_bounds__(256) void ca_prep(
    const float* __restrict__ fg, const float* __restrict__ bg,
    const float* __restrict__ mask,
    float* __restrict__ fgp, float* __restrict__ kern,
    float* __restrict__ vmat, float* __restrict__ mmv)
{
  int idx = blockIdx.x * 256 + threadIdx.x;
  if (idx >= B_N * P_TOT) return;
  int b = idx >> 12;
  int l = idx & (P_TOT - 1);
  int u = l >> 6, v = l & 63;

  // mm flag: 3x3 patch of 2x-downsampled mask sums to zero
  const float* mb = mask + (size_t)b * H_N * W_N;
  float msum = 0.f;
  for (int di = -1; di <= 1; ++di) {
    int uu = u + di; if (uu < 0 || uu >= HD) continue;
    for (int dj = -1; dj <= 1; ++dj) {
      int vv = v + dj; if (vv < 0 || vv >= HD) continue;
      msum += mb[(size_t)(2 * uu) * W_N + 2 * vv];
    }
  }
  mmv[idx] = (msum == 0.0f) ? 1.0f : 0.0f;

  const float* bgb = bg + (size_t)b * C_N * H_N * W_N;
  const float* fgb = fg + (size_t)b * C_N * H_N * W_N;

  // norm of bg (downsampled) 3x3xC patch
  float ss = 0.f;
  for (int c = 0; c < C_N; ++c) {
    const float* bc = bgb + (size_t)c * H_N * W_N;
    for (int ki = 0; ki < 3; ++ki) {
      int uu = u + ki - 1;
      for (int kj = 0; kj < 3; ++kj) {
        int vv = v + kj - 1;
        float x = (uu >= 0 && uu < HD && vv >= 0 && vv < HD)
                      ? bc[(size_t)(2 * uu) * W_N + 2 * vv] : 0.f;
        ss += x * x;
      }
    }
  }
  float inv = 1.0f / fmaxf(sqrtf(ss), NORM_EPS);

  float* kd = kern + (size_t)idx * KDIM;
  float* fd = fgp  + (size_t)idx * KDIM;
  int k = 0;
  for (int c = 0; c < C_N; ++c) {
    const float* bc = bgb + (size_t)c * H_N * W_N;
    const float* fc = fgb + (size_t)c * H_N * W_N;
    for (int ki = 0; ki < 3; ++ki) {
      int uu = u + ki - 1;
      for (int kj = 0; kj < 3; ++kj, ++k) {
        int vv = v + kj - 1;
        bool in = (uu >= 0 && uu < HD && vv >= 0 && vv < HD);
        float xb = in ? bc[(size_t)(2 * uu) * W_N + 2 * vv] : 0.f;
        float xf = in ? fc[(size_t)(2 * uu) * W_N + 2 * vv] : 0.f;
        kd[k] = xb * inv;
        fd[k] = xf;
      }
    }
  }

  // full-res 4x4 bg patches at (2u-1, 2v-1), zero-padded
  float* vd = vmat + (size_t)idx * CP;
  for (int c = 0; c < C_N; ++c) {
    const float* bc = bgb + (size_t)c * H_N * W_N;
    for (int i = 0; i < 4; ++i) {
      int yy = 2 * u - 1 + i;
      for (int j = 0; j < 4; ++j) {
        int xx = 2 * v - 1 + j;
        float x = (yy >= 0 && yy < H_N && xx >= 0 && xx < W_N)
                      ? bc[(size_t)yy * W_N + xx] : 0.f;
        vd[c * 16 + i * 4 + j] = x;
      }
    }
  }
}

// ------------- kernel 2: fused scores + masked softmax + recon GEMM -------------
// One wave (32 threads) per block; wave owns 16 p's; double-buffered async
// global->LDS staging of kern/V tiles overlapped with WMMA compute.
__global__ __launch_bounds__(32) void ca_attn(
    const float* __restrict__ fgp, const float* __restrict__ kern,
    const float* __restrict__ vmat, const float* __restrict__ mmv,
    float* __restrict__ Rws)
{
  __shared__ __align__(16) float Qs[16][KP2];      // fgp tile (16 x 144)
  __shared__ __align__(16) float Ks[2][16][KP2];   // kern tiles (double buffered)
  __shared__ __align__(16) float Vs[2][16][CP];    // V tiles (double buffered)
  __shared__ float Ws[16][17];                     // weight-tile transpose staging
  __shared__ float Ms[2][16];                      // mm values
  __shared__ float Ds[16];                         // softmax denominators

  const int b    = blockIdx.y;
  const int lane = threadIdx.x;
  const int hi   = lane >> 4;
  const int lo   = lane & 15;
  const int p0   = blockIdx.x * 16;

  // stage this wave's Q tile (one-time, synchronous)
  const float* qg = fgp + (size_t)(b * P_TOT + p0) * KDIM;
  for (int idx = lane; idx < 16 * KDIM; idx += 32)
    Qs[idx / KDIM][idx % KDIM] = qg[idx];

  const float* kbase = kern + (size_t)b * P_TOT * KDIM;
  const float* vbase = vmat + (size_t)b * P_TOT * CP;
  const float* mbase = mmv + (size_t)b * P_TOT;

  // issue staging for chunk ch into buffer bf
  auto issue = [&](int ch, int bf) {
    const float* kg = kbase + (size_t)ch * 16 * KDIM;
    for (int i = lane; i < 16 * (KDIM / 4); i += 32) {      // 576 x 16B
      int row = i / (KDIM / 4), c4 = i % (KDIM / 4);
      copy16_to_lds(kg + row * KDIM + c4 * 4, &Ks[bf][row][c4 * 4]);
    }
    const float* vg = vbase + (size_t)ch * 16 * CP;
    float* vl = &Vs[bf][0][0];
    for (int i = lane; i < (16 * CP) / 4; i += 32)          // 1024 x 16B
      copy16_to_lds(vg + i * 4, vl + i * 4);
    if (lane < 16) Ms[bf][lane] = mbase[ch * 16 + lane];
  };

  issue(0, 0);

  v8f acc[16];
  for (int nt = 0; nt < 16; ++nt)
    for (int j = 0; j < 8; ++j) acc[nt][j] = 0.0f;
  float m = -1e30f;
  float d = 0.0f;

  for (int ch = 0; ch < NCH; ++ch) {
    const int bf = ch & 1;
    wait_async_done();           // current chunk's async copies landed
    __syncthreads();             // (1-wave WG: compiler memory fence)
    if (ch + 1 < NCH) issue(ch + 1, bf ^ 1);   // overlap next copy with compute

    // ---- scores tile: D[M=l, N=p] = Ks(16xK) * Qs^T(Kx16), K=144 in 36 steps
    v8f sc;
    for (int j = 0; j < 8; ++j) sc[j] = 0.0f;
    for (int ks = 0; ks < 36; ++ks) {
      int k0 = ks * 4 + 2 * hi;              // f32 A/B layout: half-wave K pairs
      v2f a;  a.x  = Ks[bf][lo][k0]; a.y  = Ks[bf][lo][k0 + 1];
      v2f bb; bb.x = Qs[lo][k0];     bb.y = Qs[lo][k0 + 1];
      sc = __builtin_amdgcn_wmma_f32_16x16x4_f32(
              false, a, false, bb, (short)0, sc, false, false);
    }

    // ---- online masked softmax over l (per lane: p = lo, rows l = i + 8*hi)
    float z[8];
    float cmax = -1e30f;
    for (int i = 0; i < 8; ++i) {
      float mv = Ms[bf][i + 8 * hi];
      z[i] = SM_SCALE * sc[i] * mv;
      cmax = fmaxf(cmax, z[i]);
    }
    cmax = fmaxf(cmax, __shfl_xor(cmax, 16, 32));
    float nm    = fmaxf(m, cmax);
    float scale = __expf(m - nm);
    float psum  = 0.0f;
    float w[8];
    for (int i = 0; i < 8; ++i) {
      float e = __expf(z[i] - nm);
      psum += e;                             // masked l contribute exp(0-nm)
      w[i] = Ms[bf][i + 8 * hi] * e;         // attn numerator, masked -> 0
    }
    psum += __shfl_xor(psum, 16, 32);
    d = d * scale + psum;
    m = nm;

    for (int nt = 0; nt < 16; ++nt)
      for (int j = 0; j < 8; ++j) acc[nt][j] *= scale;

    // ---- transpose weights into A-layout via LDS (wave-local)
    __builtin_amdgcn_wave_barrier();
    for (int i = 0; i < 8; ++i) Ws[i + 8 * hi][lo] = w[i];
    __builtin_amdgcn_wave_barrier();

    // ---- recon: D[M=p, N=c'] += w^T(16x16) * Vs(16x256), 4 K-steps x 16 N-tiles
    for (int ks = 0; ks < 4; ++ks) {
      int lk = ks * 4 + 2 * hi;
      v2f a2; a2.x = Ws[lk][lo]; a2.y = Ws[lk + 1][lo];
      for (int nt = 0; nt < 16; ++nt) {
        v2f b2; b2.x = Vs[bf][lk][nt * 16 + lo]; b2.y = Vs[bf][lk + 1][nt * 16 + lo];
        acc[nt] = __builtin_amdgcn_wmma_f32_16x16x4_f32(
                     false, a2, false, b2, (short)0, acc[nt], false, false);
      }
    }
  }

  // ---- normalize by softmax denominator and write R[p, c']
  __builtin_amdgcn_wave_barrier();
  if (hi == 0) Ds[lo] = d;                   // both halves hold identical d
  __builtin_amdgcn_wave_barrier();
  float dl[8];
  for (int i = 0; i < 8; ++i) dl[i] = 1.0f / Ds[i + 8 * hi];

  float* rg = Rws + (size_t)(b * P_TOT + p0) * CP;
  for (int nt = 0; nt < 16; ++nt) {
    int cc = nt * 16 + lo;
    for (int i = 0; i < 8; ++i) {
      int p = i + 8 * hi;
      rg[(size_t)p * CP + cc] = acc[nt][i] * dl[i];
    }
  }
}

// ---------------- kernel 3: transposed-conv overlap gather ----------------
__global__ __launch_bounds__(256) void ca_recon(
    const float* __restrict__ Rws, float* __restrict__ out)
{
  int idx = blockIdx.x * 256 + threadIdx.x;
  if (idx >= B_N * C_N * H_N * W_N) return;
  int x = idx & 127;
  int y = (idx >> 7) & 127;
  int c = (idx >> 14) & 15;
  int b = idx >> 18;

  float s = 0.f;
  for (int i = (y + 1) & 1; i < 4; i += 2) {
    int u = (y + 1 - i) >> 1;
    if (u < 0 || u > 63) continue;
    for (int j = (x + 1) & 1; j < 4; j += 2) {
      int v = (x + 1 - j) >> 1;
      if (v < 0 || v > 63) continue;
      s += Rws[((size_t)(b * P_TOT + u * 64 + v)) * CP + c * 16 + i * 4 + j];
    }
  }
  out[idx] = 0.25f * s;
}

// ---------------- launcher ----------------
extern "C" void kernel_launch(void* const* d_in, const int* in_sizes, int n_in,
                              void* d_out, int out_size, void* d_ws, size_t ws_size,
                              hipStream_t stream) {
  const float* fg   = (const float*)d_in[0];
  const float* bg   = (const float*)d_in[1];
  const float* mask = (const float*)d_in[2];
  // d_in[3..5] = rate(2), stride(1), ksize(3): hardcoded above.

  float* ws   = (float*)d_ws;
  float* fgp  = ws;                                   // 8*4096*144
  float* kern = fgp  + (size_t)B_N * P_TOT * KDIM;    // 8*4096*144
  float* vmat = kern + (size_t)B_N * P_TOT * KDIM;    // 8*4096*256
  float* mmv  = vmat + (size_t)B_N * P_TOT * CP;      // 8*4096
  float* Rws  = mmv  + (size_t)B_N * P_TOT;           // 8*4096*256
  float* out  = (float*)d_out;

  {
    int total = B_N * P_TOT;
    ca_prep<<<(total + 255) / 256, 256, 0, stream>>>(fg, bg, mask, fgp, kern, vmat, mmv);
  }
  {
    dim3 grid(P_TOT / 16, B_N);   // 1 wave per block, 16 p per wave
    ca_attn<<<grid, 32, 0, stream>>>(fgp, kern, vmat, mmv, Rws);
  }
  {
    int total = B_N * C_N * H_N * W_N;
    ca_recon<<<(total + 255) / 256, 256, 0, stream>>>(Rws, out);
  }
}